// Qwen2_5_VLVisionAttention_54657753809296
// MI455X (gfx1250) — compile-verified
//
#include <hip/hip_runtime.h>
#include <hip/hip_bf16.h>
#include <math.h>

typedef float v2f __attribute__((ext_vector_type(2)));
typedef float v8f __attribute__((ext_vector_type(8)));

#define S_LEN 4096
#define DMODEL 1280
#define NHEADS 16
#define HDIM 80

// ---------------------------------------------------------------------------
// CDNA5 async global->LDS copy (ASYNCcnt-tracked, no VGPR round trip).
// vdst = LDS byte address (low 32 bits of the generic pointer are the LDS
// offset per the flat-aperture mapping), vaddr = 64-bit global address.
// ---------------------------------------------------------------------------
__device__ __forceinline__ void async_copy_b128(void* lds, const void* gptr) {
  const unsigned int laddr = (unsigned int)(uintptr_t)lds;
  asm volatile("global_load_async_to_lds_b128 %0, %1, off"
               :: "v"(laddr), "v"(gptr) : "memory");
}
__device__ __forceinline__ void wait_async_le8() {
  asm volatile("s_wait_asynccnt 0x8" ::: "memory");
}
__device__ __forceinline__ void wait_async_0() {
  asm volatile("s_wait_asynccnt 0x0" ::: "memory");
}

// ---------------------------------------------------------------------------
// GEMM (NT): C[M,N] = A[M,K] * B[N,K]^T + bias[N]
// 256 threads (8 waves). Tile 128x128, K-chunk 32, double-buffered LDS filled
// by async-to-LDS copies. Wave (wr,wc) computes a 32x64 strip: 2x4 fp32 WMMA
// accumulators.
// ---------------------------------------------------------------------------
__global__ __launch_bounds__(256)
void gemm_nt_wmma(const float* __restrict__ A, const float* __restrict__ B,
                  const float* __restrict__ bias, float* __restrict__ C,
                  int M, int N, int K) {
  __shared__ __align__(16) float As[2][128 * 32];
  __shared__ __align__(16) float Bs[2][128 * 32];

  const int tid  = threadIdx.x;
  const int lane = tid & 31;
  const int wv   = tid >> 5;          // 0..7
  const int wr   = wv & 3;            // wave row (4 x 32 rows)
  const int wc   = wv >> 2;           // wave col (2 x 64 cols)
  const int lm   = lane & 15;
  const int ks   = lane >> 4;         // K+0 / K+2 half select (fp32 WMMA layout)
  const int tileM = blockIdx.x * 128;
  const int tileN = blockIdx.y * 128;

  // staging assignment: 256 threads, 32 rows x 8 float4-cols per pass, 4 passes
  const int srow = tid >> 3;          // 0..31
  const int scol = (tid & 7) * 4;     // 0..28

  v8f acc[2][4] = {};

  const int nc = K / 32;

  // prologue: stage chunk 0 into buffer 0
  {
#pragma unroll
    for (int p = 0; p < 4; ++p) {
      const int r = p * 32 + srow;
      async_copy_b128(&As[0][r * 32 + scol],
                      A + (size_t)(tileM + r) * K + scol);
      async_copy_b128(&Bs[0][r * 32 + scol],
                      B + (size_t)(tileN + r) * K + scol);
    }
  }

  for (int c = 0; c < nc; ++c) {
    const int cur = c & 1;
    if (c + 1 < nc) {
      const int kc = (c + 1) * 32;
#pragma unroll
      for (int p = 0; p < 4; ++p) {
        const int r = p * 32 + srow;
        async_copy_b128(&As[cur ^ 1][r * 32 + scol],
                        A + (size_t)(tileM + r) * K + kc + scol);
        async_copy_b128(&Bs[cur ^ 1][r * 32 + scol],
                        B + (size_t)(tileN + r) * K + kc + scol);
      }
      wait_async_le8();   // in-order completion => chunk c fully resident
    } else {
      wait_async_0();
    }
    __syncthreads();

    const float* as = As[cur];
    const float* bs = Bs[cur];
#pragma unroll
    for (int kk = 0; kk < 32; kk += 4) {
      v2f a[2], b[4];
#pragma unroll
      for (int rt = 0; rt < 2; ++rt) {
        const float* p = as + (wr * 32 + rt * 16 + lm) * 32 + kk + 2 * ks;
        a[rt].x = p[0];
        a[rt].y = p[1];
      }
#pragma unroll
      for (int jt = 0; jt < 4; ++jt) {
        const float* p = bs + (wc * 64 + jt * 16 + lm) * 32 + kk + 2 * ks;
        b[jt].x = p[0];
        b[jt].y = p[1];
      }
#pragma unroll
      for (int rt = 0; rt < 2; ++rt)
#pragma unroll
        for (int jt = 0; jt < 4; ++jt)
          acc[rt][jt] = __builtin_amdgcn_wmma_f32_16x16x4_f32(
              false, a[rt], false, b[jt], (short)0, acc[rt][jt], false, false);
    }
    __syncthreads();  // reads done before this buffer is refilled (chunk c+2)
  }

  // C/D layout: VGPR r -> rows r (lanes 0-15) and r+8 (lanes 16-31), N = lane%16
#pragma unroll
  for (int rt = 0; rt < 2; ++rt) {
#pragma unroll
    for (int jt = 0; jt < 4; ++jt) {
      const int n  = tileN + wc * 64 + jt * 16 + lm;
      const float bv = bias[n];
#pragma unroll
      for (int r = 0; r < 8; ++r) {
        const int m = tileM + wr * 32 + rt * 16 + r + 8 * ks;
        C[(size_t)m * N + n] = acc[rt][jt][r] + bv;
      }
    }
  }
}

// ---------------------------------------------------------------------------
// RoPE + scatter: qkv[S,3,H,hd] -> q,k,v in [H,S,hd] (q,k rotated)
// ---------------------------------------------------------------------------
__global__ void rope_scatter(const float* __restrict__ qkv,
                             const float* __restrict__ cosb,
                             const float* __restrict__ sinb,
                             float* __restrict__ q, float* __restrict__ k,
                             float* __restrict__ v) {
  const int idx = blockIdx.x * blockDim.x + threadIdx.x;
  if (idx >= S_LEN * DMODEL) return;
  const int s = idx / DMODEL;
  const int c = idx - s * DMODEL;       // h*80 + d
  const int h = c / HDIM;
  const int d = c - h * HDIM;

  const float cv = cosb[s * HDIM + d];
  const float sv = sinb[s * HDIM + d];
  const float* row = qkv + (size_t)s * (3 * DMODEL);

  const int  pc  = (d < HDIM / 2) ? (c + HDIM / 2) : (c - HDIM / 2);
  const float sg = (d < HDIM / 2) ? -1.f : 1.f;

  const float qv = row[c];
  const float qr = sg * row[pc];
  const float kv = row[DMODEL + c];
  const float kr = sg * row[DMODEL + pc];

  const size_t o = ((size_t)h * S_LEN + s) * HDIM + d;
  q[o] = qv * cv + qr * sv;
  k[o] = kv * cv + kr * sv;
  v[o] = row[2 * DMODEL + c];
}

// ---------------------------------------------------------------------------
// Flash-style block-diagonal attention.
// grid = H * (S/64); block = 128 threads (4 waves, 16 query rows each).
// QK^T and P*V both on fp32 WMMA; online softmax with wave32 shfl reductions.
// ---------------------------------------------------------------------------
__global__ __launch_bounds__(128)
void attn_wmma(const float* __restrict__ q, const float* __restrict__ k,
               const float* __restrict__ v, const int* __restrict__ cu,
               float* __restrict__ out) {
  __shared__ __align__(16) float Qs[64 * 80];
  __shared__ __align__(16) float Ps[4][16 * 16];

  const int nqt   = S_LEN / 64;
  const int h     = blockIdx.x / nqt;
  const int qbase = (blockIdx.x % nqt) * 64;
  const int tid  = threadIdx.x;
  const int lane = tid & 31;
  const int wv   = tid >> 5;
  const int lm   = lane & 15;
  const int ks   = lane >> 4;

  // find segment containing this query tile (tiles never straddle segments)
  int seg = 0;
#pragma unroll
  for (int i = 1; i <= 3; ++i)
    if (qbase >= cu[i]) seg = i;
  const int kstart = cu[seg];
  const int kend   = cu[seg + 1];

  // stage Q tile (64 rows x 80) into LDS: 2 threads/row, 10 float4 each
  {
    const int row = tid >> 1;
    const int off = (tid & 1) * 40;
    const float* src = q + ((size_t)h * S_LEN + qbase + row) * HDIM + off;
    float* dst = Qs + row * 80 + off;
#pragma unroll
    for (int j = 0; j < 10; ++j)
      *(float4*)(dst + j * 4) = *(const float4*)(src + j * 4);
  }
  __syncthreads();

  v8f o[5] = {};
  float rmax[8], rsum[8];
#pragma unroll
  for (int r = 0; r < 8; ++r) { rmax[r] = -1e30f; rsum[r] = 0.f; }
  const float scale = rsqrtf((float)HDIM);

  const float* kh = k + (size_t)h * S_LEN * HDIM;
  const float* vh = v + (size_t)h * S_LEN * HDIM;

  for (int kb = kstart; kb < kend; kb += 16) {
    // ---- scores tile: S = Q (16x80) * K^T (80x16), 20 WMMA k-steps ----
    v8f sacc = {};
    const float* krow = kh + (size_t)(kb + lm) * HDIM;
#pragma unroll
    for (int kk = 0; kk < HDIM; kk += 4) {
      v2f a, b;
      a.x = Qs[(wv * 16 + lm) * 80 + kk + 2 * ks];
      a.y = Qs[(wv * 16 + lm) * 80 + kk + 2 * ks + 1];
      b.x = krow[kk + 2 * ks];
      b.y = krow[kk + 2 * ks + 1];
      sacc = __builtin_amdgcn_wmma_f32_16x16x4_f32(
          false, a, false, b, (short)0, sacc, false, false);
    }

    // ---- online softmax (row reductions across the 16 N-lanes) ----
    float p[8];
#pragma unroll
    for (int r = 0; r < 8; ++r) {
      const float sv = sacc[r] * scale;
      float mx = sv;
#pragma unroll
      for (int off = 8; off >= 1; off >>= 1)
        mx = fmaxf(mx, __shfl_xor(mx, off, 32));
      const float nm    = fmaxf(rmax[r], mx);
      const float alpha = __expf(rmax[r] - nm);
      const float pv    = __expf(sv - nm);
      float psum = pv;
#pragma unroll
      for (int off = 8; off >= 1; off >>= 1)
        psum += __shfl_xor(psum, off, 32);
      rsum[r] = rsum[r] * alpha + psum;
      rmax[r] = nm;
      p[r]    = pv;
#pragma unroll
      for (int j = 0; j < 5; ++j) o[j][r] *= alpha;
    }

    // ---- re-fragment P (C-layout -> A-layout) through per-wave LDS ----
    __syncthreads();
#pragma unroll
    for (int r = 0; r < 8; ++r)
      Ps[wv][(r + 8 * ks) * 16 + lm] = p[r];
    __syncthreads();

    // ---- O += P (16x16) * V (16x80), 4 k-steps x 5 d-tiles ----
#pragma unroll
    for (int kk = 0; kk < 16; kk += 4) {
      v2f a;
      a.x = Ps[wv][lm * 16 + kk + 2 * ks];
      a.y = Ps[wv][lm * 16 + kk + 2 * ks + 1];
      const float* vrow0 = vh + (size_t)(kb + kk + 2 * ks) * HDIM;
      const float* vrow1 = vh + (size_t)(kb + kk + 2 * ks + 1) * HDIM;
#pragma unroll
      for (int j = 0; j < 5; ++j) {
        v2f b;
        b.x = vrow0[j * 16 + lm];
        b.y = vrow1[j * 16 + lm];
        o[j] = __builtin_amdgcn_wmma_f32_16x16x4_f32(
            false, a, false, b, (short)0, o[j], false, false);
      }
    }
  }

  // ---- normalize and write [S, D] with D = H*hd ----
#pragma unroll
  for (int r = 0; r < 8; ++r) {
    const float inv  = 1.f / rsum[r];
    const int   srow = qbase + wv * 16 + r + 8 * ks;
#pragma unroll
    for (int j = 0; j < 5; ++j)
      out[(size_t)srow * DMODEL + h * HDIM + j * 16 + lm] = o[j][r] * inv;
  }
}

// ---------------------------------------------------------------------------
extern "C" void kernel_launch(void* const* d_in, const int* in_sizes, int n_in,
                              void* d_out, int out_size, void* d_ws, size_t ws_size,
                              hipStream_t stream) {
  const float* hidden = (const float*)d_in[0];
  const int*   cu     = (const int*)d_in[1];
  const float* cosb   = (const float*)d_in[2];
  const float* sinb   = (const float*)d_in[3];
  const float* qkv_w  = (const float*)d_in[4];
  const float* qkv_b  = (const float*)d_in[5];
  const float* proj_w = (const float*)d_in[6];
  const float* proj_b = (const float*)d_in[7];
  float* out = (float*)d_out;

  float* ws  = (float*)d_ws;
  float* qkv = ws;                                     // S * 3D, reused as attn_out
  float* q   = ws + (size_t)S_LEN * 3 * DMODEL;        // [H,S,hd]
  float* k   = q + (size_t)S_LEN * DMODEL;
  float* v   = k + (size_t)S_LEN * DMODEL;

  // 1) qkv = hidden @ qkv_w^T + qkv_b
  dim3 g1(S_LEN / 128, (3 * DMODEL) / 128);
  gemm_nt_wmma<<<g1, 256, 0, stream>>>(hidden, qkv_w, qkv_b, qkv,
                                       S_LEN, 3 * DMODEL, DMODEL);

  // 2) RoPE + scatter to head-major layout
  const int tot = S_LEN * DMODEL;
  rope_scatter<<<(tot + 255) / 256, 256, 0, stream>>>(qkv, cosb, sinb, q, k, v);

  // 3) block-diagonal flash attention -> attn_out (reuses qkv buffer)
  float* attn_out = qkv;
  attn_wmma<<<NHEADS * (S_LEN / 64), 128, 0, stream>>>(q, k, v, cu, attn_out);

  // 4) out = attn_out @ proj_w^T + proj_b
  dim3 g4(S_LEN / 128, DMODEL / 128);
  gemm_nt_wmma<<<g4, 256, 0, stream>>>(attn_out, proj_w, proj_b, out,
                                       S_LEN, DMODEL, DMODEL);
}